// RPN_loss_44547400794354
// MI455X (gfx1250) — compile-verified
//
#include <hip/hip_runtime.h>

// RPN loss for MI455X (gfx1250, wave32).
// Memory-bound reduction (~18 MB traffic -> ~1us at 23.3 TB/s HBM).
// WMMA f32 16x16x4 (B = ones) is used as the in-register wave-reduction tree.

typedef __attribute__((ext_vector_type(2))) float v2f;
typedef __attribute__((ext_vector_type(8))) float v8f;

#define HW   65536   // 256*256
#define WDIM 256

// ---- wave32 sum via V_WMMA_F32_16X16X4_F32 -------------------------------
// A[m][0] = p[lane m], A[m][2] = p[lane m+16], A[m][1]=A[m][3]=0; B = ones.
// D[m][n] = p[m] + p[m+16] for every n.  C/D layout: vgpr j, lanes 0-15 hold
// M=j, lanes 16-31 hold M=j+8.  So sum_j c[j] gives lanes 0-15: sum over
// M=0..7, lanes 16-31: sum over M=8..15; one shfl_xor(16) completes the sum.
__device__ __forceinline__ float wave_sum_wmma(float p) {
  v2f a; a.x = p;    a.y = 0.0f;
  v2f b; b.x = 1.0f; b.y = 1.0f;
  v8f c = {};
  c = __builtin_amdgcn_wmma_f32_16x16x4_f32(
      /*neg_a=*/false, a, /*neg_b=*/false, b,
      /*c_mod=*/(short)0, c, /*reuse_a=*/false, /*reuse_b=*/false);
  float s = ((c[0] + c[1]) + (c[2] + c[3])) + ((c[4] + c[5]) + (c[6] + c[7]));
  s += __shfl_xor(s, 16);   // all 32 lanes now hold the wave total
  return s;
}

// Block reduction: wave totals -> LDS -> thread 0 sums in fixed order
// (deterministic), writes *dst = total * scale.
__device__ __forceinline__ void block_partial_to(float v, float* lds,
                                                 float* dst, float scale) {
  float w = wave_sum_wmma(v);
  int wave = threadIdx.x >> 5;
  if ((threadIdx.x & 31) == 0) lds[wave] = w;
  __syncthreads();
  if (threadIdx.x == 0) {
    int nw = blockDim.x >> 5;
    float t = 0.0f;
    for (int i = 0; i < nw; ++i) t += lds[i];
    *dst = t * scale;
  }
  __syncthreads();
}

__device__ __forceinline__ float softplus_f(float x) {
  // stable: max(x,0) + log1p(exp(-|x|))
  return fmaxf(x, 0.0f) + log1pf(__expf(-fabsf(x)));
}

__device__ __forceinline__ float smooth_l1(float d) {
  float ad = fabsf(d);
  return (ad < 1.0f) ? 0.5f * d * d : ad - 0.5f;
}

// ---- objectness loss: sum over 2 pred planes (c16,c17) vs target (c0,c1) --
// 131072 threads, one float4 per plane pair each -> no tail, EXEC stays full.
__global__ void __launch_bounds__(256)
obj_loss_kernel(const float* __restrict__ pred, const float* __restrict__ tgt,
                float* __restrict__ part) {
  __shared__ float lds[8];
  const int HW4 = HW / 4;                       // 16384 = 2^14
  int i = blockIdx.x * 256 + threadIdx.x;       // 0 .. 131071
  int b = i >> 14;
  int j = i & (HW4 - 1);
  const float4* p16 = (const float4*)(pred + ((size_t)b * 54 + 16) * HW);
  const float4* p17 = (const float4*)(pred + ((size_t)b * 54 + 17) * HW);
  const float4* t0  = (const float4*)(tgt  + ((size_t)b * 6  + 0) * HW);
  const float4* t1  = (const float4*)(tgt  + ((size_t)b * 6  + 1) * HW);
  float4 pa = p16[j], pb = p17[j], ta = t0[j], tb = t1[j];
  float acc = 0.0f;
  acc += softplus_f(pa.x) - pa.x * ta.x;
  acc += softplus_f(pa.y) - pa.y * ta.y;
  acc += softplus_f(pa.z) - pa.z * ta.z;
  acc += softplus_f(pa.w) - pa.w * ta.w;
  acc += softplus_f(pb.x) - pb.x * tb.x;
  acc += softplus_f(pb.y) - pb.y * tb.y;
  acc += softplus_f(pb.z) - pb.z * tb.z;
  acc += softplus_f(pb.w) - pb.w * tb.w;
  block_partial_to(acc, lds, part + blockIdx.x, 1.0f);
}

// ---- box loss: one thread per (b, n); 64 blocks x 256 = 16384, no tail ----
__global__ void __launch_bounds__(256)
box_loss_kernel(const float* __restrict__ pred, const float* __restrict__ tgt,
                const int* __restrict__ loc, float* __restrict__ part) {
  __shared__ float lds[8];
  // anchors: for s in {8,16,32}: for r in {0.5,1,2}: w=16*s/sqrt(r), h=16*s*sqrt(r)
  const float inv_w[9] = {
      0.0055242717280199034f, 0.0078125f,   0.0110485434560398068f,
      0.0027621358640099517f, 0.00390625f,  0.0055242717280199034f,
      0.0013810679320049758f, 0.001953125f, 0.0027621358640099517f};
  const float inv_h[9] = {
      0.0110485434560398068f, 0.0078125f,   0.0055242717280199034f,
      0.0055242717280199034f, 0.00390625f,  0.0027621358640099517f,
      0.0027621358640099517f, 0.001953125f, 0.0013810679320049758f};
  int t = blockIdx.x * 256 + threadIdx.x;   // 0 .. 16383
  int b = t >> 11;                          // batch
  int n = t & 2047;                         // location index
  int x = loc[2 * n], y = loc[2 * n + 1];
  int off = x * WDIM + y;
  const float* tb = tgt + ((size_t)b * 6 + 2) * HW + off;
  float tg0 = tb[0], tg1 = tb[HW], tg2 = tb[2 * HW], tg3 = tb[3 * HW];
  float lt2 = __logf(tg2), lt3 = __logf(tg3);   // log(t/w)-log(p/w)=log(t)-log(p)
  const float* pbase = pred + ((size_t)b * 54 + 18) * HW + off;
  float acc = 0.0f;
#pragma unroll
  for (int a = 0; a < 9; ++a) {
    const float* pa = pbase + (size_t)(a * 4) * HW;
    float p0 = pa[0], p1 = pa[HW], p2 = pa[2 * HW], p3 = pa[3 * HW];
    float d0 = (p0 - tg0) * inv_w[a];
    float d1 = (p1 - tg1) * inv_h[a];
    float d2 = __logf(p2) - lt2;
    float d3 = __logf(p3) - lt3;
    acc += smooth_l1(d0) + smooth_l1(d1) + smooth_l1(d2) + smooth_l1(d3);
  }
  block_partial_to(acc, lds, part + 512 + blockIdx.x, 1.0f);
}

// ---- final: one block sums 512 obj partials and 64 box partials ----------
__global__ void __launch_bounds__(256)
final_kernel(const float* __restrict__ part, float* __restrict__ out) {
  __shared__ float lds[8];
  float a = 0.0f;
  for (int i = threadIdx.x; i < 512; i += 256) a += part[i];
  block_partial_to(a, lds, out + 0, 1.0f / 9.0f);          // loss_obj / NUM_ANCHORS
  float bsum = (threadIdx.x < 64) ? part[512 + threadIdx.x] : 0.0f;
  block_partial_to(bsum, lds, out + 1, 1.0f);              // loss_box
}

extern "C" void kernel_launch(void* const* d_in, const int* in_sizes, int n_in,
                              void* d_out, int out_size, void* d_ws, size_t ws_size,
                              hipStream_t stream) {
  const float* pred = (const float*)d_in[0];   // (8, 54, 256, 256) f32
  const float* tgt  = (const float*)d_in[1];   // (8, 6, 256, 256)  f32
  const int*   loc  = (const int*)d_in[2];     // (2048, 2)         i32
  float* out  = (float*)d_out;                 // {loss_obj, loss_box}
  float* part = (float*)d_ws;                  // 512 + 64 partials

  obj_loss_kernel<<<512, 256, 0, stream>>>(pred, tgt, part);
  box_loss_kernel<<<64, 256, 0, stream>>>(pred, tgt, loc, part);
  final_kernel<<<1, 256, 0, stream>>>(part, out);
}